// RoMoAligner_87883620811554
// MI455X (gfx1250) — compile-verified
//
#include <hip/hip_runtime.h>
#include <cstdint>

// Batched ragged linear resample (resample_mel_embeddings) for MI455X/gfx1250.
// Bandwidth-bound gather (~50 MB total traffic -> ~2.1 us at 23.3 TB/s).
// Strategy: wave32-per-output-frame, async global->LDS row staging
// (global_load_async_to_lds_b128 + s_wait_asynccnt), lerp in VALU,
// non-temporal b128 stores for the streamed output.

typedef float v4f __attribute__((ext_vector_type(4)));

#define NB 32
#define NI 256
#define NJ 2048
#define NC 512

// ---------------------------------------------------------------------------
// Kernel 1: per-sample valid lengths T (mel) and L (text) from byte masks.
// jnp bool arrays are 1 byte per element.
// ---------------------------------------------------------------------------
__global__ __launch_bounds__(256) void RoMo_lengths_kernel(
    const unsigned char* __restrict__ text_mask,
    const unsigned char* __restrict__ mel_mask,
    int2* __restrict__ lens) {
  const int b = blockIdx.x;
  const int t = threadIdx.x;
  __shared__ int sT, sL;
  if (t == 0) { sT = 0; sL = 0; }
  __syncthreads();
  int l = text_mask[b * NI + t] ? 1 : 0;          // NI == blockDim.x
  int m = 0;
#pragma unroll
  for (int k = 0; k < NJ / 256; ++k)
    m += mel_mask[b * NJ + k * 256 + t] ? 1 : 0;
  atomicAdd(&sL, l);
  atomicAdd(&sT, m);
  __syncthreads();
  if (t == 0) lens[b] = make_int2(sT, sL);
}

// ---------------------------------------------------------------------------
// Kernel 2: one wave32 per output frame. 8 frames (8 waves) per block.
// grid.x = NB * (NI/8). LDS: per wave, two staged source rows (2 x 2048 B).
// ---------------------------------------------------------------------------
__global__ __launch_bounds__(256) void RoMo_resample_kernel(
    const float* __restrict__ mel,      // (B, J, C) f32
    const int2* __restrict__ lens,      // (B,) {T, L}
    float* __restrict__ out) {          // (B, I, C) f32
  __shared__ __align__(16) float smem[8 * 1024];  // 32 KB: 8 waves x 2 rows x 512 f32

  const int wave = threadIdx.x >> 5;
  const int lane = threadIdx.x & 31;
  const int b    = blockIdx.x >> 5;    // 32 frame-tiles per sample
  const int tile = blockIdx.x & 31;
  const int i    = tile * 8 + wave;    // output frame index, wave-uniform

  const int2 tl = lens[b];
  const int T = tl.x;
  const int L = tl.y;

  v4f* __restrict__ orow = (v4f*)(out + ((size_t)b * NI + i) * NC);

  if (i >= L) {                         // wave-uniform branch: EXEC stays full
    const v4f z = (v4f)0.0f;
#pragma unroll
    for (int k = 0; k < 4; ++k)
      __builtin_nontemporal_store(z, orow + k * 32 + lane);
    return;
  }

  // align_corners=False sampling grid (matches F.interpolate linear)
  const float Tf = (float)T;
  const float Lf = (float)(L > 0 ? L : 1);
  float pos = ((float)i + 0.5f) * (Tf / Lf) - 0.5f;
  pos = fminf(fmaxf(pos, 0.0f), Tf - 1.0f);
  const int   i0 = (int)pos;            // pos >= 0 -> trunc == floor
  const int   i1 = (i0 + 1 < T) ? (i0 + 1) : (T - 1);
  const float w  = pos - (float)i0;

  const float* __restrict__ g0 = mel + ((size_t)b * NJ + (size_t)i0) * NC;
  const float* __restrict__ g1 = mel + ((size_t)b * NJ + (size_t)i1) * NC;

  // Wave-private LDS staging region; low 32 bits of a shared generic pointer
  // are the LDS byte offset (LDS aperture tag lives in the high 32 bits).
  float* sw = &smem[wave * 1024];
  const unsigned ldsBase = (unsigned)(uintptr_t)sw;

  // Stage both source rows: 8x async b128 per lane-group, direct to LDS.
  // No data VGPRs consumed -> minimal register pressure, max occupancy.
#pragma unroll
  for (int k = 0; k < 4; ++k) {
    const unsigned l0 = ldsBase + (unsigned)(k * 512 + lane * 16);
    const unsigned l1 = l0 + 2048u;
    const float* a0 = g0 + k * 128 + lane * 4;
    const float* a1 = g1 + k * 128 + lane * 4;
    asm volatile("global_load_async_to_lds_b128 %0, %1, off"
                 :: "v"(l0), "v"(a0) : "memory");
    asm volatile("global_load_async_to_lds_b128 %0, %1, off"
                 :: "v"(l1), "v"(a1) : "memory");
  }
  // Wave reads only the LDS it wrote: ASYNCcnt wait suffices, no barrier.
  asm volatile("s_wait_asynccnt 0x0" ::: "memory");

  // Lerp from LDS (ds_load_b128) and stream out with NT b128 stores.
#pragma unroll
  for (int k = 0; k < 4; ++k) {
    const v4f a = *(const v4f*)&sw[k * 128 + lane * 4];
    const v4f c = *(const v4f*)&sw[512 + k * 128 + lane * 4];
    const v4f r = a + w * (c - a);      // (1-w)*g0 + w*g1
    __builtin_nontemporal_store(r, orow + k * 32 + lane);
  }
}

// ---------------------------------------------------------------------------
// Host launcher. Inputs (setup_inputs order):
//   d_in[0] mel_embeddings  (B,J,C) f32
//   d_in[1] durations_normalized (unused by the reference body)
//   d_in[2] text_mask (B,I) bool (1 byte/elem)
//   d_in[3] mel_mask  (B,J) bool (1 byte/elem)
// d_out: (B,I,C) f32. d_ws: 32 int2 for lengths.
// ---------------------------------------------------------------------------
extern "C" void kernel_launch(void* const* d_in, const int* in_sizes, int n_in,
                              void* d_out, int out_size, void* d_ws, size_t ws_size,
                              hipStream_t stream) {
  const float* mel = (const float*)d_in[0];
  const unsigned char* text_mask = (const unsigned char*)d_in[2];
  const unsigned char* mel_mask  = (const unsigned char*)d_in[3];
  int2* lens = (int2*)d_ws;

  RoMo_lengths_kernel<<<NB, 256, 0, stream>>>(text_mask, mel_mask, lens);
  RoMo_resample_kernel<<<NB * (NI / 8), 256, 0, stream>>>(mel, lens, (float*)d_out);
}